// ConvDeIndexer_72756745994786
// MI455X (gfx1250) — compile-verified
//
#include <hip/hip_runtime.h>

// MI455X / gfx1250, wave32. Fused de-index + stride-4 transposed conv.
// Formulated as (16 positions) x (K=64 channels) x (N=512 = co*kh*kw) GEMM
// per workgroup tile using V_WMMA_F32_16X16X4_F32 chains (fp32-exact; the
// op is output-bandwidth bound: 256 MB stores ~= 11 us at 23.3 TB/s).
// Output stores + single-use input loads are non-temporal so the 256 MB
// stream does not evict the L2-resident 1 MB weight matrix (re-read by all
// 8192 workgroups, ~1 GB cumulative L2 traffic).

typedef __attribute__((ext_vector_type(2))) float v2f;
typedef __attribute__((ext_vector_type(8))) float v8f;

#define C_CODE  64
#define K_TOP   4
#define M_GRP   8
#define BATCH   4
#define WDIM    64
#define HDIM    64
#define XPITCH  68              // 64 channels + 4 pad -> conflict-free A fetch

__global__ __launch_bounds__(256)
void deindex_deconv_wmma(const int*   __restrict__ ind_pos,
                         const int*   __restrict__ ind_neg,
                         const float* __restrict__ val_pos,
                         const float* __restrict__ val_neg,
                         const float* __restrict__ weight,   // (512,32,4,4) = per-group 64x512 row-major
                         float*       __restrict__ out)      // (4,256,256,256)
{
    __shared__ float X[16 * XPITCH];    // dense A tile: 16 positions x 64 channels

    const int tid  = threadIdx.x;
    const int blk  = blockIdx.x;        // 8192 blocks = b(4) * g(8) * row(64) * colt(4)
    const int colt = blk & 3;
    const int row  = (blk >> 2) & 63;
    const int g    = (blk >> 8) & 7;
    const int b    = blk >> 11;
    const int col0 = colt * 16;

    // --- zero the dense tile ---
    #pragma unroll
    for (int i = tid; i < 16 * XPITCH; i += 256) X[i] = 0.0f;
    __syncthreads();

    // --- deterministic scatter: one thread per position, pos first then neg ---
    if (tid < 16) {
        const int p    = tid;
        const int base = ((b * K_TOP * M_GRP + g) * WDIM + row) * HDIM + col0 + p; // k=0
        const int kstr = M_GRP * WDIM * HDIM;                                      // k stride
        float* xr = &X[p * XPITCH];
        #pragma unroll
        for (int k = 0; k < K_TOP; ++k) {
            const int idx = __builtin_nontemporal_load(ind_pos + base + k * kstr);
            xr[idx]       = __builtin_nontemporal_load(val_pos + base + k * kstr);
        }
        #pragma unroll
        for (int k = 0; k < K_TOP; ++k) {         // neg overrides pos (reference .set order)
            const int idx = __builtin_nontemporal_load(ind_neg + base + k * kstr);
            xr[idx]       = __builtin_nontemporal_load(val_neg + base + k * kstr);
        }
    }
    __syncthreads();

    const int lane  = tid & 31;
    const int wave  = tid >> 5;
    const int mrow  = lane & 15;   // A: M row / B,D: N column
    const int hl    = lane >> 4;   // lane half selects K pair (A/B) and M half (D)

    // --- A fragments for all 16 K-steps, held in VGPRs, reused by 4 N-tiles ---
    v2f afrag[16];
    #pragma unroll
    for (int ks = 0; ks < 16; ++ks)
        afrag[ks] = *(const v2f*)&X[mrow * XPITCH + ks * 4 + hl * 2];

    // --- per-group weight matrix: 64 x 512 row-major, L2-resident (default RT loads) ---
    const float* Bg = weight + (size_t)g * (C_CODE * 512);
    const int    nlane = lane & 15;

    for (int nt = 0; nt < 4; ++nt) {
        const int co = wave * 4 + nt;                 // N-tile index == output channel in group
        const float* bp = Bg + (hl * 2) * 512 + co * 16 + nlane;

        v8f acc = {0.f, 0.f, 0.f, 0.f, 0.f, 0.f, 0.f, 0.f};
        #pragma unroll
        for (int ks = 0; ks < 16; ++ks) {             // K = 64 in steps of 4
            v2f bfrag;
            bfrag.x = bp[ks * 2048];                  // row ks*4 + 2*hl
            bfrag.y = bp[ks * 2048 + 512];            // row ks*4 + 2*hl + 1
            acc = __builtin_amdgcn_wmma_f32_16x16x4_f32(
                      false, afrag[ks], false, bfrag, (short)0, acc, false, false);
        }

        // --- scatter-store D: lane holds n = co*16 + nlane -> (kh,kw); VGPR v -> position ---
        const int kh   = nlane >> 2;
        const int kw   = nlane & 3;
        const int orow = row * 4 + kh;
        float* op = out + (((size_t)(b * 256 + g * 32 + co) * 256 + orow) * 256)
                        + col0 * 4 + kw;
        #pragma unroll
        for (int v = 0; v < 8; ++v) {
            const int p = hl * 8 + v;                 // M = v (lanes 0-15) or v+8 (lanes 16-31)
            __builtin_nontemporal_store(acc[v], op + p * 4);   // TH=NT: stream past L2
        }
    }
}

extern "C" void kernel_launch(void* const* d_in, const int* in_sizes, int n_in,
                              void* d_out, int out_size, void* d_ws, size_t ws_size,
                              hipStream_t stream) {
    const int*   ind_pos = (const int*)  d_in[0];
    const int*   ind_neg = (const int*)  d_in[1];
    const float* val_pos = (const float*)d_in[2];
    const float* val_neg = (const float*)d_in[3];
    const float* weight  = (const float*)d_in[4];
    float*       out     = (float*)d_out;

    dim3 grid(BATCH * M_GRP * WDIM * (HDIM / 16));   // 8192 workgroups
    dim3 block(256);                                  // 8 waves (wave32)
    hipLaunchKernelGGL(deindex_deconv_wmma, grid, block, 0, stream,
                       ind_pos, ind_neg, val_pos, val_neg, weight, out);
}